// WarpMesh_84224308675189
// MI455X (gfx1250) — compile-verified
//
#include <hip/hip_runtime.h>
#include <hip/hip_bf16.h>

typedef __attribute__((ext_vector_type(2))) float v2f;
typedef __attribute__((ext_vector_type(8))) float v8f;

#define W_IMG 384
#define H_IMG 384
#define NFACE 450
#define NFACE_PAD 464   /* 29 tiles of 16 faces */
#define NTILE 29
#define TOL 1e-7f
#define EPS 1e-8f

// ---------------------------------------------------------------------------
// Precompute: per-face edge-function coefficients with 1/area folded in, plus
// normalized per-face UVs.  Degenerate / padding faces get cc = -1e30 so the
// WMMA result is always < -tol (never "inside").
// ---------------------------------------------------------------------------
__global__ void wm_precompute(const float* __restrict__ points,
                              const float* __restrict__ uvp,
                              const int*   __restrict__ faces,
                              float2* __restrict__ cxy,   // [3][NFACE_PAD] (cx,cy)
                              float*  __restrict__ ccc,   // [3][NFACE_PAD] cc
                              float2* __restrict__ fuv)   // [3][NFACE_PAD] uv verts
{
    int f = blockIdx.x * blockDim.x + threadIdx.x;
    if (f >= NFACE_PAD) return;

    const float inv_wh = 1.0f / 383.0f;        // 1/(W-1)
    const float mxy    = 383.0f / 384.0f;      // (W-1)/W

    float v0x = 0.f, v0y = 0.f, v1x = 0.f, v1y = 0.f, v2x = 0.f, v2y = 0.f;
    float u0x = 0.f, u0y = 0.f, u1x = 0.f, u1y = 0.f, u2x = 0.f, u2y = 0.f;
    bool realf = (f < NFACE);
    if (realf) {
        int i0 = faces[f * 3 + 0];
        int i1 = faces[f * 3 + 1];
        int i2 = faces[f * 3 + 2];
        v0x = (points[i0 * 2 + 0] * (2.0f * inv_wh) - 1.0f) * mxy;
        v0y = (points[i0 * 2 + 1] * (2.0f * inv_wh) - 1.0f) * mxy;
        v1x = (points[i1 * 2 + 0] * (2.0f * inv_wh) - 1.0f) * mxy;
        v1y = (points[i1 * 2 + 1] * (2.0f * inv_wh) - 1.0f) * mxy;
        v2x = (points[i2 * 2 + 0] * (2.0f * inv_wh) - 1.0f) * mxy;
        v2y = (points[i2 * 2 + 1] * (2.0f * inv_wh) - 1.0f) * mxy;
        u0x = uvp[i0 * 2 + 0] * inv_wh;  u0y = uvp[i0 * 2 + 1] * inv_wh;
        u1x = uvp[i1 * 2 + 0] * inv_wh;  u1y = uvp[i1 * 2 + 1] * inv_wh;
        u2x = uvp[i2 * 2 + 0] * inv_wh;  u2y = uvp[i2 * 2 + 1] * inv_wh;
    }

    float area  = (v1x - v0x) * (v2y - v0y) - (v1y - v0y) * (v2x - v0x);
    bool  valid = realf && (fabsf(area) >= EPS);
    float inv   = valid ? (1.0f / area) : 0.0f;

    // edge_fn(a,b)(p) = cx*px + cy*py + cc, cx=-(by-ay), cy=(bx-ax), cc=ax*by-ay*bx
    float cx0 = -(v2y - v1y) * inv, cy0 = (v2x - v1x) * inv, cc0 = (v1x * v2y - v1y * v2x) * inv;
    float cx1 = -(v0y - v2y) * inv, cy1 = (v0x - v2x) * inv, cc1 = (v2x * v0y - v2y * v0x) * inv;
    float cx2 = -(v1y - v0y) * inv, cy2 = (v1x - v0x) * inv, cc2 = (v0x * v1y - v0y * v1x) * inv;
    if (!valid) {
        cx0 = cy0 = cx1 = cy1 = cx2 = cy2 = 0.0f;
        cc0 = cc1 = cc2 = -1e30f;
    }

    cxy[0 * NFACE_PAD + f] = make_float2(cx0, cy0);
    cxy[1 * NFACE_PAD + f] = make_float2(cx1, cy1);
    cxy[2 * NFACE_PAD + f] = make_float2(cx2, cy2);
    ccc[0 * NFACE_PAD + f] = cc0;
    ccc[1 * NFACE_PAD + f] = cc1;
    ccc[2 * NFACE_PAD + f] = cc2;
    fuv[0 * NFACE_PAD + f] = make_float2(u0x, u0y);
    fuv[1 * NFACE_PAD + f] = make_float2(u1x, u1y);
    fuv[2 * NFACE_PAD + f] = make_float2(u2x, u2y);
}

// ---------------------------------------------------------------------------
// Raster kernel.  One wave32 handles 16 consecutive pixels (one row segment).
// Edge evaluation: D[16px x 16faces] = A[16px x 4] * B[4 x 16faces] per edge
// via V_WMMA_F32_16X16X4_F32.  A lanes 0-15 carry K0/K1 = (px,py), lanes
// 16-31 carry K2/K3 = (1,0); B lanes 0-15 carry (cx,cy), lanes 16-31 (cc,0).
// Only the winning face index is carried through the loop / reduction; the
// barycentrics are re-evaluated once for the winner (6 FMAs) at the end.
// ---------------------------------------------------------------------------
__global__ void __launch_bounds__(256) wm_raster(
    const float*  __restrict__ img,
    const float2* __restrict__ cxy,
    const float*  __restrict__ ccc,
    const float2* __restrict__ fuv,
    float*        __restrict__ out)
{
    const int lane = threadIdx.x & 31;
    const int wave = threadIdx.x >> 5;
    const int sub  = lane & 15;
    const int half = lane >> 4;
    const int pixBase = (blockIdx.x * 8 + wave) * 16;

    // Build A (16x4 f32 matrix, striped per ISA layout).
    const int   pm  = pixBase + sub;
    const float pxv = ((float)(pm % W_IMG) + 0.5f) * (2.0f / W_IMG) - 1.0f;
    const float pyv = ((float)(pm / W_IMG) + 0.5f) * (2.0f / H_IMG) - 1.0f;
    v2f a;
    if (lane < 16) { a.x = pxv;  a.y = pyv;  }   // K=0,1
    else           { a.x = 1.0f; a.y = 0.0f; }   // K=2,3

    const v8f zero = {};
    int bf[8];
#pragma unroll
    for (int r = 0; r < 8; ++r) bf[r] = 0x7fffffff;

    for (int t = 0; t < NTILE; ++t) {
        const int faceN = t * 16 + sub;
        if (t + 1 < NTILE) {
            __builtin_prefetch(&cxy[0 * NFACE_PAD + (t + 1) * 16], 0, 1);   // global_prefetch_b8
            __builtin_prefetch(&ccc[0 * NFACE_PAD + (t + 1) * 16], 0, 1);
        }
        v2f b0, b1, b2;
        if (lane < 16) {
            float2 q0 = cxy[0 * NFACE_PAD + faceN];
            float2 q1 = cxy[1 * NFACE_PAD + faceN];
            float2 q2 = cxy[2 * NFACE_PAD + faceN];
            b0.x = q0.x; b0.y = q0.y;
            b1.x = q1.x; b1.y = q1.y;
            b2.x = q2.x; b2.y = q2.y;
        } else {
            b0.x = ccc[0 * NFACE_PAD + faceN]; b0.y = 0.0f;
            b1.x = ccc[1 * NFACE_PAD + faceN]; b1.y = 0.0f;
            b2.x = ccc[2 * NFACE_PAD + faceN]; b2.y = 0.0f;
        }

        v8f d0 = __builtin_amdgcn_wmma_f32_16x16x4_f32(false, a, false, b0, (short)0, zero, false, false);
        v8f d1 = __builtin_amdgcn_wmma_f32_16x16x4_f32(false, a, false, b1, (short)0, zero, false, false);
        v8f d2 = __builtin_amdgcn_wmma_f32_16x16x4_f32(false, a, false, b2, (short)0, zero, false, false);

#pragma unroll
        for (int r = 0; r < 8; ++r) {
            bool ins  = (d0[r] >= -TOL) & (d1[r] >= -TOL) & (d2[r] >= -TOL);
            bool take = ins && (bf[r] == 0x7fffffff);   // first (lowest) face wins
            bf[r] = take ? faceN : bf[r];
        }
    }

    // Min-face butterfly across the 16 lanes of each half (faces live on lanes).
#pragma unroll
    for (int m = 1; m <= 8; m <<= 1) {
#pragma unroll
        for (int r = 0; r < 8; ++r) {
            int of = __shfl_xor(bf[r], m, 32);
            bf[r] = (of < bf[r]) ? of : bf[r];
        }
    }

    // D layout: VGPR r holds pixel row (r + 8*half).  Lanes with sub<8 each
    // emit one pixel: pixel = pixBase + half*8 + sub, taken from register sub.
    if (sub < 8) {
        int face = bf[0];
#pragma unroll
        for (int r = 1; r < 8; ++r) face = (sub == r) ? bf[r] : face;

        const int p = pixBase + half * 8 + sub;
        float c0 = 0.f, c1 = 0.f, c2 = 0.f;
        if (face != 0x7fffffff) {
            // Re-evaluate the winner's barycentrics: same affine form the WMMA used.
            const float qx = ((float)(p % W_IMG) + 0.5f) * (2.0f / W_IMG) - 1.0f;
            const float qy = ((float)(p / W_IMG) + 0.5f) * (2.0f / H_IMG) - 1.0f;
            float2 e0 = cxy[0 * NFACE_PAD + face]; float k0 = ccc[0 * NFACE_PAD + face];
            float2 e1 = cxy[1 * NFACE_PAD + face]; float k1 = ccc[1 * NFACE_PAD + face];
            float2 e2 = cxy[2 * NFACE_PAD + face]; float k2 = ccc[2 * NFACE_PAD + face];
            float s0 = fmaf(e0.x, qx, fmaf(e0.y, qy, k0));
            float s1 = fmaf(e1.x, qx, fmaf(e1.y, qy, k1));
            float s2 = fmaf(e2.x, qx, fmaf(e2.y, qy, k2));

            float2 u0 = fuv[0 * NFACE_PAD + face];
            float2 u1 = fuv[1 * NFACE_PAD + face];
            float2 u2 = fuv[2 * NFACE_PAD + face];
            float ux = s0 * u0.x + s1 * u1.x + s2 * u2.x;
            float uy = s0 * u0.y + s1 * u1.y + s2 * u2.y;
            float x = fminf(fmaxf(ux * 383.0f, 0.0f), 383.0f);
            float y = fminf(fmaxf(uy * 383.0f, 0.0f), 383.0f);
            int x0 = (int)floorf(x), y0 = (int)floorf(y);
            int x1 = min(x0 + 1, 383), y1 = min(y0 + 1, 383);
            float fx = x - (float)x0, fy = y - (float)y0;
            const float* p00 = img + (y0 * W_IMG + x0) * 3;
            const float* p01 = img + (y0 * W_IMG + x1) * 3;
            const float* p10 = img + (y1 * W_IMG + x0) * 3;
            const float* p11 = img + (y1 * W_IMG + x1) * 3;
            float w00 = (1.f - fx) * (1.f - fy), w01 = fx * (1.f - fy);
            float w10 = (1.f - fx) * fy,         w11 = fx * fy;
            c0 = p00[0] * w00 + p01[0] * w01 + p10[0] * w10 + p11[0] * w11;
            c1 = p00[1] * w00 + p01[1] * w01 + p10[1] * w10 + p11[1] * w11;
            c2 = p00[2] * w00 + p01[2] * w01 + p10[2] * w10 + p11[2] * w11;
        }
        out[p * 3 + 0] = c0;
        out[p * 3 + 1] = c1;
        out[p * 3 + 2] = c2;
    }
}

extern "C" void kernel_launch(void* const* d_in, const int* in_sizes, int n_in,
                              void* d_out, int out_size, void* d_ws, size_t ws_size,
                              hipStream_t stream) {
    const float* points = (const float*)d_in[0];   // 256 x 2 f32
    const float* uvp    = (const float*)d_in[1];   // 256 x 2 f32
    const float* img    = (const float*)d_in[2];   // 1 x 384 x 384 x 3 f32
    const int*   faces  = (const int*)d_in[3];     // 450 x 3 i32
    float*       out    = (float*)d_out;           // 384 x 384 x 3 f32

    float2* cxy = (float2*)d_ws;                        // 3*464 float2
    float*  ccc = (float*)(cxy + 3 * NFACE_PAD);        // 3*464 float
    float2* fuv = (float2*)(ccc + 3 * NFACE_PAD);       // 3*464 float2

    wm_precompute<<<2, 256, 0, stream>>>(points, uvp, faces, cxy, ccc, fuv);

    const int pixels = W_IMG * H_IMG;                   // 147456
    const int blocks = pixels / (8 * 16);               // 8 waves/block, 16 px/wave
    wm_raster<<<blocks, 256, 0, stream>>>(img, cxy, ccc, fuv, out);
}